// FeatureOctreeBase_82901458747492
// MI455X (gfx1250) — compile-verified
//
#include <hip/hip_runtime.h>

#ifndef __has_builtin
#define __has_builtin(x) 0
#endif

#define THREADS 256

typedef float v4f __attribute__((ext_vector_type(4)));
typedef int   v4i __attribute__((ext_vector_type(4)));

#if __has_builtin(__builtin_amdgcn_global_load_async_to_lds_b128)
#define USE_ASYNC_IDX 1
// Builtin expects v4i pointers: arg0 in addrspace(1) (prints as __device__),
// arg1 in addrspace(3). Launder through integers (flat->LDS is addr[31:0]
// truncation per ISA 10.2, so the integer cast matches hardware semantics).
typedef __attribute__((address_space(1))) v4i gv4i;
typedef __attribute__((address_space(3))) v4i sv4i;
#define ASYNC_CP_B128(gsrc, ldst)                                   \
    __builtin_amdgcn_global_load_async_to_lds_b128(                 \
        (gv4i*)(unsigned long long)(const void*)(gsrc),             \
        (sv4i*)(unsigned long long)(void*)(ldst), 0, 0)
#else
#define USE_ASYNC_IDX 0
#endif

__device__ __forceinline__ float poly_smooth(float c) {
    float d = c - floorf(c);
    return d * d * (3.0f - 2.0f * d);   // 3d^2 - 2d^3
}

// acc[f] += sum_k p[k] * table[idx[k]][f], with row (szm1) forced to zero.
__device__ __forceinline__ void accum_level(
    const float* __restrict__ table, int szm1,
    v4i ia, v4i ib,
    float bx, float by, float bz, float scale,
    float* acc)
{
    float tx = poly_smooth(scale * bx);
    float ty = poly_smooth(scale * by);
    float tz = poly_smooth(scale * bz);
    float wx[2] = {1.0f - tx, tx};
    float wy[2] = {1.0f - ty, ty};
    float wz[2] = {1.0f - tz, tz};

    int id[8];
    id[0] = ia.x; id[1] = ia.y; id[2] = ia.z; id[3] = ia.w;
    id[4] = ib.x; id[5] = ib.y; id[6] = ib.z; id[7] = ib.w;

    // Issue all 16 b128 gathers first for memory-level parallelism.
    v4f r0[8], r1[8];
#pragma unroll
    for (int k = 0; k < 8; ++k) {
        const v4f* row = reinterpret_cast<const v4f*>(table)
                         + (size_t)(unsigned)id[k] * 2;
        r0[k] = row[0];
        r1[k] = row[1];
    }
#pragma unroll
    for (int k = 0; k < 8; ++k) {
        float p = wx[(k >> 2) & 1] * wy[(k >> 1) & 1] * wz[k & 1];
        p = (id[k] == szm1) ? 0.0f : p;   // reference zeroes the last table row
        acc[0] = fmaf(p, r0[k].x, acc[0]);
        acc[1] = fmaf(p, r0[k].y, acc[1]);
        acc[2] = fmaf(p, r0[k].z, acc[2]);
        acc[3] = fmaf(p, r0[k].w, acc[3]);
        acc[4] = fmaf(p, r1[k].x, acc[4]);
        acc[5] = fmaf(p, r1[k].y, acc[5]);
        acc[6] = fmaf(p, r1[k].z, acc[6]);
        acc[7] = fmaf(p, r1[k].w, acc[7]);
    }
}

__global__ __launch_bounds__(THREADS) void feature_octree_gather_kernel(
    const float* __restrict__ coord,
    const float* __restrict__ tblA,   // level 12: table2 (largest)
    const float* __restrict__ tblB,   // level 11: table1
    const float* __restrict__ tblC,   // level 10: table0
    const int* __restrict__ idxA,     // idx0
    const int* __restrict__ idxB,     // idx1
    const int* __restrict__ idxC,     // idx2
    float* __restrict__ out,
    int n, int szAm1, int szBm1, int szCm1)
{
    const int tid = threadIdx.x;
    const int q = blockIdx.x * THREADS + tid;

#if USE_ASYNC_IDX
    // Stage each lane's three 32B index rows into LDS with CDNA5 async
    // loads (ASYNCcnt), overlapping the copies with coord load + weights.
    __shared__ __align__(16) int lidx[3][THREADS * 8];
    if (q < n) {
        const size_t go = (size_t)q * 8;
        const int lo = tid * 8;
        ASYNC_CP_B128(idxA + go,     &lidx[0][lo]);
        ASYNC_CP_B128(idxA + go + 4, &lidx[0][lo + 4]);
        ASYNC_CP_B128(idxB + go,     &lidx[1][lo]);
        ASYNC_CP_B128(idxB + go + 4, &lidx[1][lo + 4]);
        ASYNC_CP_B128(idxC + go,     &lidx[2][lo]);
        ASYNC_CP_B128(idxC + go + 4, &lidx[2][lo + 4]);
    }
#endif
    if (q >= n) return;

    // Streaming coord read: non-temporal so it never evicts table lines
    // from the 192MB L2 (tables ~90MB are L2-resident).
    const float* cp = coord + (size_t)q * 3;
    const float bx = __builtin_nontemporal_load(cp + 0) * 0.5f + 0.5f;
    const float by = __builtin_nontemporal_load(cp + 1) * 0.5f + 0.5f;
    const float bz = __builtin_nontemporal_load(cp + 2) * 0.5f + 0.5f;

    float acc[8] = {0.f, 0.f, 0.f, 0.f, 0.f, 0.f, 0.f, 0.f};

#if USE_ASYNC_IDX
#if __has_builtin(__builtin_amdgcn_s_wait_asynccnt)
    __builtin_amdgcn_s_wait_asynccnt(0);
#else
    asm volatile("s_wait_asynccnt 0" ::: "memory");
#endif
    asm volatile("" ::: "memory");  // fence LDS reads below the wait
    const int lo = tid * 8;
    v4i iA0 = *(const v4i*)&lidx[0][lo];
    v4i iA1 = *(const v4i*)&lidx[0][lo + 4];
    v4i iB0 = *(const v4i*)&lidx[1][lo];
    v4i iB1 = *(const v4i*)&lidx[1][lo + 4];
    v4i iC0 = *(const v4i*)&lidx[2][lo];
    v4i iC1 = *(const v4i*)&lidx[2][lo + 4];
#else
    const size_t go = (size_t)q * 8;
    v4i iA0 = __builtin_nontemporal_load((const v4i*)(idxA + go));
    v4i iA1 = __builtin_nontemporal_load((const v4i*)(idxA + go + 4));
    v4i iB0 = __builtin_nontemporal_load((const v4i*)(idxB + go));
    v4i iB1 = __builtin_nontemporal_load((const v4i*)(idxB + go + 4));
    v4i iC0 = __builtin_nontemporal_load((const v4i*)(idxC + go));
    v4i iC1 = __builtin_nontemporal_load((const v4i*)(idxC + go + 4));
#endif

    accum_level(tblA, szAm1, iA0, iA1, bx, by, bz, 4096.0f, acc);  // level 12
    accum_level(tblB, szBm1, iB0, iB1, bx, by, bz, 2048.0f, acc);  // level 11
    accum_level(tblC, szCm1, iC0, iC1, bx, by, bz, 1024.0f, acc);  // level 10

    // Non-temporal streaming store of the result.
    v4f o0, o1;
    o0.x = acc[0]; o0.y = acc[1]; o0.z = acc[2]; o0.w = acc[3];
    o1.x = acc[4]; o1.y = acc[5]; o1.z = acc[6]; o1.w = acc[7];
    v4f* op = reinterpret_cast<v4f*>(out + (size_t)q * 8);
    __builtin_nontemporal_store(o0, op);
    __builtin_nontemporal_store(o1, op + 1);
}

extern "C" void kernel_launch(void* const* d_in, const int* in_sizes, int n_in,
                              void* d_out, int out_size, void* d_ws, size_t ws_size,
                              hipStream_t stream) {
    const float* coord  = (const float*)d_in[0];
    const float* table0 = (const float*)d_in[1];  // 200001 x 8
    const float* table1 = (const float*)d_in[2];  // 600001 x 8
    const float* table2 = (const float*)d_in[3];  // 2000001 x 8
    const int*   idx0   = (const int*)d_in[4];    // pairs with table2 (level 12)
    const int*   idx1   = (const int*)d_in[5];    // pairs with table1 (level 11)
    const int*   idx2   = (const int*)d_in[6];    // pairs with table0 (level 10)

    const int n   = in_sizes[0] / 3;
    const int szA = in_sizes[3] / 8;  // rows in table2
    const int szB = in_sizes[2] / 8;  // rows in table1
    const int szC = in_sizes[1] / 8;  // rows in table0

    const int blocks = (n + THREADS - 1) / THREADS;
    hipLaunchKernelGGL(feature_octree_gather_kernel,
                       dim3(blocks), dim3(THREADS), 0, stream,
                       coord, table2, table1, table0,
                       idx0, idx1, idx2,
                       (float*)d_out, n, szA - 1, szB - 1, szC - 1);
}